// EdgeConv_reorg_67508295958886
// MI455X (gfx1250) — compile-verified
//
#include <hip/hip_runtime.h>
#include <hip/hip_bf16.h>

typedef __attribute__((ext_vector_type(2))) float v2f;
typedef __attribute__((ext_vector_type(8))) float v8f;
typedef __attribute__((ext_vector_type(4))) unsigned int u32x4;
typedef __attribute__((ext_vector_type(8))) int i32x8;
typedef __attribute__((ext_vector_type(4))) int i32x4;

#define IN_FEAT 64
#define OUT_FEAT 64
#define LDS_W_STRIDE 68   // 64 + 4 pad floats: bank-conflict-free B reads (4n+4t covers all 64 banks)

// ---------------------------------------------------------------------------
// TDM: load a row-major 64x64 f32 matrix from global into LDS with +4-DWORD
// padding after every 64-DWORD row (-> LDS row stride 68 floats).
// D# per CDNA5 ISA Ch.8: group0 {count=1, lds_addr, global_addr, type=2},
// group1 {data_size=4B, pad_enable, pad_interval=5 (64 dw), pad_amount=3 (4 dw),
//         tensor_dim0/1=64, tile_dim0/1=64, tensor_dim0_stride=64}.
// This toolchain exposes the 6-arg builtin: (g0, g1, g2, g3, g4, cpol).
// ---------------------------------------------------------------------------
#if defined(__gfx1250__) && __has_builtin(__builtin_amdgcn_tensor_load_to_lds)
#define HAVE_TDM 1
__device__ __forceinline__ void tdm_load_w64x64(const float* gsrc, unsigned lds_byte_off)
{
    unsigned long long ga = (unsigned long long)(const void*)gsrc;
    u32x4 g0;
    g0[0] = 1u;                                           // count=1, user mode, no gather
    g0[1] = lds_byte_off;                                 // lds_addr (bytes)
    g0[2] = (unsigned)(ga & 0xFFFFFFFFu);                 // global_addr[31:0]
    g0[3] = (unsigned)((ga >> 32) & 0x01FFFFFFu)          // global_addr[56:32]
          | (2u << 30);                                   // type=2 ("image")
    i32x8 g1;
    g1[0] = (2 << 16)                                     // data_size = 4 bytes
          | (1 << 20)                                     // pad_enable
          | (5 << 22)                                     // pad_interval: 64 DWORDs
          | (3 << 25);                                    // pad_amount: 4 DWORDs
    g1[1] = (64 & 0xFFFF) << 16;                          // tensor_dim0 lo16 = 64
    g1[2] = ((64 & 0xFFFF) << 16);                        // dim0 hi16=0 | tensor_dim1 lo16 = 64
    g1[3] = (64 << 16);                                   // dim1 hi16=0 | tile_dim0 = 64
    g1[4] = 64;                                           // tile_dim1 = 64, tile_dim2 = 0
    g1[5] = 64;                                           // tensor_dim0_stride lo32 = 64
    g1[6] = 0;                                            // stride0 hi16 | stride1 lo16
    g1[7] = 0;
    i32x4 z4 = {0, 0, 0, 0};                              // groups 2/3 unused (2D tensor)
    i32x8 z8 = {0, 0, 0, 0, 0, 0, 0, 0};
    __builtin_amdgcn_tensor_load_to_lds(g0, g1, z4, z4, z8, 0);
}
#else
#define HAVE_TDM 0
#endif

// ---------------------------------------------------------------------------
// Kernel 1: projection. Each wave computes a 16-row x 64-col tile of both
// h_theta = feat @ theta_w.T + theta_b and diff = (feat @ phi_w.T + phi_b) - h_theta
// using V_WMMA_F32_16X16X4_F32 (f32-exact path), B-fragments from LDS.
//
// A (16x4 f32), ISA 7.12.2: lanes 0-15 -> M=lane, K=4t+{0,1};
//                           lanes 16-31 -> M=lane-16, K=4t+{2,3}
// B (4x16 f32): lanes 0-15 -> N=lane, K=4t+{0,1}; lanes 16-31 -> N=lane-16, K=4t+{2,3}
// C/D (16x16 f32): VGPR r: lanes 0-15 -> M=r, lanes 16-31 -> M=r+8; N=lane&15
// ---------------------------------------------------------------------------
__global__ __launch_bounds__(256) void edgeconv_proj_kernel(
    const float* __restrict__ feat,
    const float* __restrict__ theta_w, const float* __restrict__ theta_b,
    const float* __restrict__ phi_w,   const float* __restrict__ phi_b,
    float* __restrict__ h_theta, float* __restrict__ diff, int n_nodes)
{
    __shared__ float wT[64 * LDS_W_STRIDE];   // theta_w staged, padded rows
    __shared__ float wP[64 * LDS_W_STRIDE];   // phi_w   staged, padded rows

    // ---- stage weights into LDS (once per block) ----
#if HAVE_TDM
    if (threadIdx.x < 32) {                   // one wave issues both TDM DMAs
        tdm_load_w64x64(theta_w, (unsigned)(size_t)&wT[0]);
        tdm_load_w64x64(phi_w,   (unsigned)(size_t)&wP[0]);
        __builtin_amdgcn_s_wait_tensorcnt(0); // data in LDS before barrier signal
    }
#else
    for (int idx = threadIdx.x; idx < 64 * 64; idx += 256) {
        const int n = idx >> 6, k = idx & 63;
        wT[n * LDS_W_STRIDE + k] = theta_w[idx];
        wP[n * LDS_W_STRIDE + k] = phi_w[idx];
    }
#endif
    __syncthreads();                          // every wave reaches this (no early return)

    const int lane    = threadIdx.x & 31;
    const int waveId  = (blockIdx.x * blockDim.x + threadIdx.x) >> 5;
    const int rowBase = waveId * 16;

    if (rowBase < n_nodes) {                  // wave-uniform: EXEC all-1s inside
        const int m     = lane & 15;
        const int kHalf = (lane >> 4) * 2;    // 0 for lanes 0-15, 2 for lanes 16-31

        int row = rowBase + m;
        if (row >= n_nodes) row = n_nodes - 1;   // clamp loads; stores predicated later

        v8f accT[4], accP[4];
#pragma unroll
        for (int nt = 0; nt < 4; ++nt) { accT[nt] = (v8f)0.0f; accP[nt] = (v8f)0.0f; }

#pragma unroll
        for (int t = 0; t < IN_FEAT / 4; ++t) {
            const int k = t * 4 + kHalf;
            v2f a = *(const v2f*)(feat + (size_t)row * IN_FEAT + k);
#pragma unroll
            for (int nt = 0; nt < 4; ++nt) {
                const int n = nt * 16 + m;
                // B[k][n] = W[n][k]; LDS row stride 68 -> conflict-free ds_load_b64
                v2f bt = *(const v2f*)(&wT[n * LDS_W_STRIDE + k]);
                v2f bp = *(const v2f*)(&wP[n * LDS_W_STRIDE + k]);
                accT[nt] = __builtin_amdgcn_wmma_f32_16x16x4_f32(
                    false, a, false, bt, (short)0, accT[nt], false, false);
                accP[nt] = __builtin_amdgcn_wmma_f32_16x16x4_f32(
                    false, a, false, bp, (short)0, accP[nt], false, false);
            }
        }

        const int mOff = (lane >> 4) * 8;     // C/D row offset: 0 or 8
#pragma unroll
        for (int nt = 0; nt < 4; ++nt) {
            const int n  = nt * 16 + m;
            const float tb = theta_b[n];
            const float pb = phi_b[n];
#pragma unroll
            for (int r = 0; r < 8; ++r) {
                const int orow = rowBase + mOff + r;
                if (orow < n_nodes) {
                    const float ht = accT[nt][r] + tb;
                    const float hp = accP[nt][r] + pb;
                    h_theta[(size_t)orow * OUT_FEAT + n] = ht;
                    diff   [(size_t)orow * OUT_FEAT + n] = hp - ht;
                }
            }
        }
    }
}

// ---------------------------------------------------------------------------
// Kernel 2: init output to -inf (segment_max identity; re-done every call).
// ---------------------------------------------------------------------------
__global__ __launch_bounds__(256) void edgeconv_init_kernel(float* __restrict__ out, int n)
{
    const int i = blockIdx.x * blockDim.x + threadIdx.x;
    if (i < n) out[i] = -__builtin_inff();
}

// ---------------------------------------------------------------------------
// Kernel 3: one wave per edge; each lane handles 2 consecutive features.
// e = h_theta[src] + diff[dst]; atomic float max into out[dst].
// h_theta/diff/out (~77 MB) stay resident in the 192 MB L2.
// ---------------------------------------------------------------------------
__global__ __launch_bounds__(256) void edgeconv_edge_kernel(
    const int* __restrict__ src, const int* __restrict__ dst,
    const float* __restrict__ h_theta, const float* __restrict__ diff,
    float* __restrict__ out, int n_edges)
{
    const int lane = threadIdx.x & 31;
    const int e    = (blockIdx.x * blockDim.x + threadIdx.x) >> 5;
    if (e >= n_edges) return;

    const int s = src[e];
    const int d = dst[e];
    const int f = lane * 2;

    v2f a = *(const v2f*)(h_theta + (size_t)s * OUT_FEAT + f);
    v2f b = *(const v2f*)(diff    + (size_t)d * OUT_FEAT + f);
    const float e0 = a.x + b.x;
    const float e1 = a.y + b.y;

    float* p = out + (size_t)d * OUT_FEAT + f;
    __hip_atomic_fetch_max(p,     e0, __ATOMIC_RELAXED, __HIP_MEMORY_SCOPE_AGENT);
    __hip_atomic_fetch_max(p + 1, e1, __ATOMIC_RELAXED, __HIP_MEMORY_SCOPE_AGENT);
}

// ---------------------------------------------------------------------------
// Kernel 4: zero-in-degree fill: non-finite (-inf) -> 0, matching reference.
// ---------------------------------------------------------------------------
__global__ __launch_bounds__(256) void edgeconv_final_kernel(float* __restrict__ out, int n)
{
    const int i = blockIdx.x * blockDim.x + threadIdx.x;
    if (i < n) {
        const float v = out[i];
        out[i] = __builtin_isfinite(v) ? v : 0.0f;
    }
}

extern "C" void kernel_launch(void* const* d_in, const int* in_sizes, int n_in,
                              void* d_out, int out_size, void* d_ws, size_t ws_size,
                              hipStream_t stream)
{
    const float* feat    = (const float*)d_in[0];
    const int*   src     = (const int*)  d_in[1];
    const int*   dst     = (const int*)  d_in[2];
    const float* theta_w = (const float*)d_in[3];
    const float* theta_b = (const float*)d_in[4];
    const float* phi_w   = (const float*)d_in[5];
    const float* phi_b   = (const float*)d_in[6];
    float* out = (float*)d_out;

    const int n_nodes = in_sizes[0] / IN_FEAT;
    const int n_edges = in_sizes[1];

    float* h_theta = (float*)d_ws;
    float* diff    = h_theta + (size_t)n_nodes * OUT_FEAT;

    // 1) projections: 8 waves/block * 16 rows/wave = 128 rows/block
    const int blocksProj = (n_nodes + 127) / 128;
    edgeconv_proj_kernel<<<blocksProj, 256, 0, stream>>>(
        feat, theta_w, theta_b, phi_w, phi_b, h_theta, diff, n_nodes);

    // 2) init out to -inf
    const int nOut = n_nodes * OUT_FEAT;
    edgeconv_init_kernel<<<(nOut + 255) / 256, 256, 0, stream>>>(out, nOut);

    // 3) per-edge scatter-max: 8 edges (waves) per 256-thread block
    const int blocksEdge = (n_edges + 7) / 8;
    edgeconv_edge_kernel<<<blocksEdge, 256, 0, stream>>>(
        src, dst, h_theta, diff, out, n_edges);

    // 4) -inf -> 0
    edgeconv_final_kernel<<<(nOut + 255) / 256, 256, 0, stream>>>(out, nOut);
}